// LightGCN_41059887350329
// MI455X (gfx1250) — compile-verified
//
#include <hip/hip_runtime.h>
#include <hip/hip_bf16.h>

#define LG_NUSERS 100000
#define LG_NITEMS 50000
#define LG_NNODES 150000
#define LG_DIM    64
#define LG_BATCH  4096
#define LG_LAYERS 3

typedef __attribute__((ext_vector_type(2))) float v2f;
typedef __attribute__((ext_vector_type(8))) float v8f;

// ---------------------------------------------------------------------------
// SpMM scatter: h_next[dst] += val * h[src], random COO.
// Half-wave (16 lanes) per edge; each lane owns a float4 (4 of 64 dims).
// Edge arrays are streamed once per layer -> non-temporal loads so the
// 38.4MB embedding state stays resident in the 192MB L2.
// Scatter uses relaxed agent-scope f32 atomics (no-return form in hardware).
// ---------------------------------------------------------------------------
__global__ __launch_bounds__(256) void lg_spmm_scatter(
    const int* __restrict__ esrc, const int* __restrict__ edst,
    const float* __restrict__ eval, const float* __restrict__ h,
    float* __restrict__ out, int nnz)
{
    int t = blockIdx.x * blockDim.x + threadIdx.x;
    int e = t >> 4;
    if (e >= nnz) return;
    int l = t & 15;

    int   s = __builtin_nontemporal_load(esrc + e);
    int   d = __builtin_nontemporal_load(edst + e);
    float v = __builtin_nontemporal_load(eval + e);

    const float4* hp = (const float4*)(h + (size_t)s * LG_DIM) + l;
    float4 x = *hp;

    float* op = out + (size_t)d * LG_DIM + (size_t)l * 4;
    __hip_atomic_fetch_add(op + 0, v * x.x, __ATOMIC_RELAXED, __HIP_MEMORY_SCOPE_AGENT);
    __hip_atomic_fetch_add(op + 1, v * x.y, __ATOMIC_RELAXED, __HIP_MEMORY_SCOPE_AGENT);
    __hip_atomic_fetch_add(op + 2, v * x.z, __ATOMIC_RELAXED, __HIP_MEMORY_SCOPE_AGENT);
    __hip_atomic_fetch_add(op + 3, v * x.w, __ATOMIC_RELAXED, __HIP_MEMORY_SCOPE_AGENT);
}

// ---------------------------------------------------------------------------
// Gather rows mat[idx[b] + rowOff] into per-batch accumulator (set or add).
// Half-wave per batch row, float4 per lane.
// ---------------------------------------------------------------------------
__global__ __launch_bounds__(256) void lg_gather_rows(
    const int* __restrict__ idx, const float* __restrict__ mat, int rowOff,
    float* __restrict__ acc, int batch, int doAdd)
{
    int t = blockIdx.x * blockDim.x + threadIdx.x;
    int b = t >> 4;
    if (b >= batch) return;
    int l = t & 15;

    int row = idx[b] + rowOff;
    const float4* sp = (const float4*)(mat + (size_t)row * LG_DIM) + l;
    float4 x = *sp;

    float4* ap = (float4*)(acc + (size_t)b * LG_DIM) + l;
    if (doAdd) {
        float4 a = *ap;
        x.x += a.x; x.y += a.y; x.z += a.z; x.w += a.w;
    }
    *ap = x;
}

// ---------------------------------------------------------------------------
// Batched dot products on the CDNA5 matrix path.
// One wave (32 lanes) handles 16 (user,item) pairs via V_WMMA_F32_16X16X4_F32:
//   A = 16x64 user rows, B = 64x16 item rows (B[k][n] = items[n][k]),
//   D = A*B accumulated over 16 chunks of K=4; diag(D) are the dot products.
// ISA layouts (CDNA5 ISA):
//   A 16x4 f32 : lane = M in 0..15 twice; VGPR v, group g -> K = 2*g + v
//   B 4x16 f32 : lane = N in 0..15 twice; VGPR v, group g -> K = 2*g + v
//   -> both fragments load the same (row = lane&15, col = k0 + 2*(lane>>4))
//   C/D 16x16  : VGPR r: lanes 0-15 -> (M=r, N=lane); lanes 16-31 -> (M=r+8).
//   diag: lanes 0..7 -> m=lane, vgpr=lane ; lanes 24..31 -> m=lane-16, vgpr=lane-24
// EXEC is all ones before the WMMA (block = exactly one full wave, no
// divergence until the final write), as the ISA requires.
// gamma = dot(acc_u, acc_i) / 16  (the (acc/4)·(acc/4) fold).
// ---------------------------------------------------------------------------
__global__ __launch_bounds__(32) void lg_dot_wmma(
    const float* __restrict__ au, const float* __restrict__ ai,
    float* __restrict__ gamma)
{
    int pairBase = blockIdx.x * 16;
    int lane = threadIdx.x;          // 0..31
    int r  = lane & 15;              // row within the 16-pair tile
    int kg = (lane >> 4) * 2;        // K sub-offset: 0 for lanes 0-15, 2 for 16-31

    const float* up = au + (size_t)(pairBase + r) * LG_DIM;
    const float* ip = ai + (size_t)(pairBase + r) * LG_DIM;

    v8f c = {};
#pragma unroll
    for (int k0 = 0; k0 < LG_DIM; k0 += 4) {
        v2f a, b;
        a.x = up[k0 + kg];
        a.y = up[k0 + kg + 1];
        b.x = ip[k0 + kg];
        b.y = ip[k0 + kg + 1];
        // 8 args: (neg_a, A, neg_b, B, c_mod, C, reuse_a, reuse_b)
        c = __builtin_amdgcn_wmma_f32_16x16x4_f32(
            false, a, false, b, (short)0, c, false, false);
    }

    // Diagonal extraction: only 16 of 32 lanes hold a diagonal element.
    int m = -1, vi = 0;
    if (lane < 8)       { m = lane;      vi = lane;      }
    else if (lane >= 24){ m = lane - 16; vi = lane - 24; }
    if (m >= 0) {
        float dv = c[vi];
        gamma[pairBase + m] = dv * 0.0625f;   // 1/16
    }
}

// ---------------------------------------------------------------------------
// kernel_launch
// Workspace layout (floats):
//   hA  : NNODES*64   (38.4 MB)  ping
//   hB  : NNODES*64   (38.4 MB)  pong
//   au  : BATCH*64    (1 MB)     accumulated user rows
//   ai  : BATCH*64    (1 MB)     accumulated item rows
// Total ~78.9 MB — L2-resident on MI455X (192 MB).
// ---------------------------------------------------------------------------
extern "C" void kernel_launch(void* const* d_in, const int* in_sizes, int n_in,
                              void* d_out, int out_size, void* d_ws, size_t ws_size,
                              hipStream_t stream)
{
    (void)n_in; (void)out_size; (void)ws_size;

    const int*   users    = (const int*)  d_in[0];
    const int*   items    = (const int*)  d_in[1];
    const float* user_emb = (const float*)d_in[2];
    const float* item_emb = (const float*)d_in[3];
    const int*   edge_src = (const int*)  d_in[4];
    const int*   edge_dst = (const int*)  d_in[5];
    const float* edge_val = (const float*)d_in[6];
    float*       gamma    = (float*)d_out;

    const int nnz = in_sizes[4];

    float* hA = (float*)d_ws;
    float* hB = hA + (size_t)LG_NNODES * LG_DIM;
    float* au = hB + (size_t)LG_NNODES * LG_DIM;
    float* ai = au + (size_t)LG_BATCH * LG_DIM;

    // h0 = concat(user_emb, item_emb)
    hipMemcpyAsync(hA, user_emb, sizeof(float) * (size_t)LG_NUSERS * LG_DIM,
                   hipMemcpyDeviceToDevice, stream);
    hipMemcpyAsync(hA + (size_t)LG_NUSERS * LG_DIM, item_emb,
                   sizeof(float) * (size_t)LG_NITEMS * LG_DIM,
                   hipMemcpyDeviceToDevice, stream);

    // acc_u = user_emb[users] ; acc_i = item_emb[items]  (layer-0 term)
    {
        int threads = LG_BATCH * 16;
        int blocks  = (threads + 255) / 256;
        lg_gather_rows<<<blocks, 256, 0, stream>>>(users, user_emb, 0, au, LG_BATCH, 0);
        lg_gather_rows<<<blocks, 256, 0, stream>>>(items, item_emb, 0, ai, LG_BATCH, 0);
    }

    float* cur = hA;
    float* nxt = hB;
    for (int layer = 0; layer < LG_LAYERS; ++layer) {
        hipMemsetAsync(nxt, 0, sizeof(float) * (size_t)LG_NNODES * LG_DIM, stream);

        {
            long long threads = (long long)nnz * 16;
            int blocks = (int)((threads + 255) / 256);
            lg_spmm_scatter<<<blocks, 256, 0, stream>>>(
                edge_src, edge_dst, edge_val, cur, nxt, nnz);
        }

        {
            int threads = LG_BATCH * 16;
            int blocks  = (threads + 255) / 256;
            lg_gather_rows<<<blocks, 256, 0, stream>>>(users, nxt, 0,          au, LG_BATCH, 1);
            lg_gather_rows<<<blocks, 256, 0, stream>>>(items, nxt, LG_NUSERS,  ai, LG_BATCH, 1);
        }

        float* tmp = cur; cur = nxt; nxt = tmp;
    }

    // gamma[b] = dot(au[b], ai[b]) / 16 via V_WMMA_F32_16X16X4_F32
    lg_dot_wmma<<<LG_BATCH / 16, 32, 0, stream>>>(au, ai, gamma);
}